// HeteroGraphConv_66202625900919
// MI455X (gfx1250) — compile-verified
//
#include <hip/hip_runtime.h>
#include <hip/hip_bf16.h>

typedef __attribute__((ext_vector_type(2))) float v2f;
typedef __attribute__((ext_vector_type(8))) float v8f;

#define NNODES 100000
#define NEDGES 1600000
#define IN_F   512
#define OUT_F  64
#define MTILES (NNODES / 16)   // 6250 exactly

// ---------------------------------------------------------------------------
// Kernel 0: zero msum accumulator (d_out) and degree accumulator (ws)
// ---------------------------------------------------------------------------
__global__ void gc_zero_kernel(float* __restrict__ out, float* __restrict__ deg) {
    int idx = blockIdx.x * blockDim.x + threadIdx.x;
    if (idx < NNODES * OUT_F) out[idx] = 0.0f;
    if (idx < NNODES)         deg[idx] = 0.0f;
}

// ---------------------------------------------------------------------------
// Kernel 1: h = feat @ W  via V_WMMA_F32_16X16X4_F32 (fp32 WMMA, wave32)
// One wave computes one 16-row M-tile across all 4 N-tiles (N=64).
// W (512x64 = 128KB) is staged per-block into LDS, pre-swizzled into the
// per-lane B-fragment layout so the inner loop is one ds_load_b64 per frag.
// ---------------------------------------------------------------------------
__global__ void gc_gemm_kernel(const float* __restrict__ feat,
                               const float* __restrict__ W,
                               float* __restrict__ h) {
    extern __shared__ float ldsW[]; // IN_F*OUT_F floats, fragment-swizzled

    // Stage + swizzle weight: element (k,n) -> frag slot for k-step s=k/4,
    // n-tile t=n/16, lane L = ((k%4)>>1)*16 + (n%16), vgpr v = (k%4)&1.
    for (int i = threadIdx.x; i < IN_F * OUT_F; i += blockDim.x) {
        int k = i >> 6, n = i & 63;
        int s = k >> 2, km = k & 3;
        int L = ((km >> 1) << 4) | (n & 15);
        int t = n >> 4;
        ldsW[(s << 8) + (t << 6) + (L << 1) + (km & 1)] = W[i];
    }
    __syncthreads();

    int wave = threadIdx.x >> 5;
    int lane = threadIdx.x & 31;
    int mtile = blockIdx.x * 8 + wave;
    if (mtile >= MTILES) return;

    // A-fragment source: lane L -> row mtile*16 + (L%16), K base = 2*(L/16)
    int row  = (mtile << 4) + (lane & 15);
    int koff = (lane >> 4) << 1;
    const float* arow = feat + (size_t)row * IN_F + koff;
    const float* lw   = ldsW + (lane << 1);

    v8f acc0 = {}, acc1 = {}, acc2 = {}, acc3 = {};

    for (int s = 0; s < IN_F / 4; ++s) {
        v2f a  = *(const v2f*)(arow + (s << 2));
        v2f b0 = *(const v2f*)(lw + (s << 8));
        v2f b1 = *(const v2f*)(lw + (s << 8) + 64);
        v2f b2 = *(const v2f*)(lw + (s << 8) + 128);
        v2f b3 = *(const v2f*)(lw + (s << 8) + 192);
        acc0 = __builtin_amdgcn_wmma_f32_16x16x4_f32(false, a, false, b0, (short)0, acc0, false, false);
        acc1 = __builtin_amdgcn_wmma_f32_16x16x4_f32(false, a, false, b1, (short)0, acc1, false, false);
        acc2 = __builtin_amdgcn_wmma_f32_16x16x4_f32(false, a, false, b2, (short)0, acc2, false, false);
        acc3 = __builtin_amdgcn_wmma_f32_16x16x4_f32(false, a, false, b3, (short)0, acc3, false, false);
    }

    // C/D layout: vgpr r, lane L -> row = r + 8*(L/16), col = L%16
    int mbase = (mtile << 4) + ((lane >> 4) << 3);
    int col   = lane & 15;
    float* hp = h + (size_t)mbase * OUT_F + col;
#pragma unroll
    for (int r = 0; r < 8; ++r) {
        hp[(size_t)r * OUT_F +  0] = acc0[r];
        hp[(size_t)r * OUT_F + 16] = acc1[r];
        hp[(size_t)r * OUT_F + 32] = acc2[r];
        hp[(size_t)r * OUT_F + 48] = acc3[r];
    }
}

// ---------------------------------------------------------------------------
// Kernel 2: edge scatter. One wave per edge (grid-stride over waves).
// Edge index is wave-uniform (readfirstlane) -> edge loads scalarize.
// Lane i handles output features {2i, 2i+1}; fp32 atomics land in L2
// (msum is 25.6MB, resident in the 192MB L2).
// ---------------------------------------------------------------------------
__global__ void gc_edge_kernel(const float* __restrict__ h,
                               const float* __restrict__ edge_w,
                               const int*   __restrict__ esrc,
                               const int*   __restrict__ edst,
                               float* __restrict__ out,
                               float* __restrict__ deg) {
    int lane = threadIdx.x & 31;
    int wid = __builtin_amdgcn_readfirstlane(
        blockIdx.x * (blockDim.x >> 5) + (threadIdx.x >> 5));
    int nwaves = gridDim.x * (blockDim.x >> 5);

    for (int e = wid; e < NEDGES; e += nwaves) {
        int   s = esrc[e];
        int   d = edst[e];
        float w = edge_w[e];
        v2f hv = *(const v2f*)(h + (size_t)s * OUT_F + (lane << 1));
        float* op = out + (size_t)d * OUT_F + (lane << 1);
        unsafeAtomicAdd(op,     hv.x * w);
        unsafeAtomicAdd(op + 1, hv.y * w);
        if (lane == 0) unsafeAtomicAdd(deg + d, 1.0f);
    }
}

// ---------------------------------------------------------------------------
// Kernel 3: finalize in place: out = relu((deg>0 ? msum/deg : 0) + bias)
// ---------------------------------------------------------------------------
__global__ void gc_finalize_kernel(float* __restrict__ out,
                                   const float* __restrict__ deg,
                                   const float* __restrict__ bias) {
    int idx = blockIdx.x * blockDim.x + threadIdx.x;
    if (idx >= NNODES * OUT_F) return;
    int n = idx >> 6, f = idx & 63;
    float dg = deg[n];
    float v  = (dg > 0.0f) ? out[idx] / dg : 0.0f;
    v += bias[f];
    out[idx] = fmaxf(v, 0.0f);
}

// ---------------------------------------------------------------------------
extern "C" void kernel_launch(void* const* d_in, const int* in_sizes, int n_in,
                              void* d_out, int out_size, void* d_ws, size_t ws_size,
                              hipStream_t stream) {
    const float* feat   = (const float*)d_in[0];
    const float* edge_w = (const float*)d_in[1];
    const float* weight = (const float*)d_in[2];
    const float* bias   = (const float*)d_in[3];
    const int*   esrc   = (const int*)d_in[4];
    const int*   edst   = (const int*)d_in[5];
    float* out = (float*)d_out;

    float* h   = (float*)d_ws;                 // NNODES*OUT_F floats (25.6 MB)
    float* deg = h + (size_t)NNODES * OUT_F;   // NNODES floats (0.4 MB)

    // 0: zero accumulators
    {
        int total = NNODES * OUT_F;
        gc_zero_kernel<<<(total + 255) / 256, 256, 0, stream>>>(out, deg);
    }
    // 1: WMMA GEMM h = feat @ W  (dynamic LDS: full 128KB weight stage)
    {
        int blocks = (MTILES + 7) / 8;
        gc_gemm_kernel<<<blocks, 256, IN_F * OUT_F * sizeof(float), stream>>>(feat, weight, h);
    }
    // 2: edge gather/scale/scatter-add
    {
        gc_edge_kernel<<<4096, 256, 0, stream>>>(h, edge_w, esrc, edst, out, deg);
    }
    // 3: mean + bias + relu in place
    {
        int total = NNODES * OUT_F;
        gc_finalize_kernel<<<(total + 255) / 256, 256, 0, stream>>>(out, deg, bias);
    }
}